// EquivSetGNN_g_28166395527446
// MI455X (gfx1250) — compile-verified
//
#include <hip/hip_runtime.h>
#include <hip/hip_bf16.h>

// Problem constants (match reference)
#define NN_NODES 100000
#define EE_EDGES 50000
#define NNZ_INC  1280000
#define FT_DIM   128
#define HID_DIM  64
#define NCLS_DIM 32
#define NGR_DIM  256

typedef float v2f __attribute__((ext_vector_type(2)));
typedef float v8f __attribute__((ext_vector_type(8)));

// ---------------------------------------------------------------------------
// Tiled GEMM: out[M x NOUT] = epi( rowscale(A)[M x K] @ W[K x NOUT] )
// via V_WMMA_F32_16X16X4_F32 (fp32 A/B, fp32 accum).
// 128 threads = 4 waves; each wave computes a 16-row x NOUT tile.
// RSMODE: 0 = none, 1 = mask(deg>0), 2 = (deg>0 ? 1/deg : 0)
// ---------------------------------------------------------------------------
template<int K, int NOUT, bool RELU, bool BIAS, bool ACCUM, int RSMODE>
__global__ __launch_bounds__(128)
void gemm_wmma_kernel(const float* __restrict__ A,
                      const float* __restrict__ W,
                      const float* __restrict__ bias,
                      const float* __restrict__ deg,
                      float* __restrict__ out, int M) {
  __shared__ float sW[K * NOUT];
  const int tid = threadIdx.x;
  for (int i = tid; i < K * NOUT; i += 128) sW[i] = W[i];
  __syncthreads();

  const int wid  = tid >> 5;
  const int lane = tid & 31;
  const int half = lane >> 4;   // 0: lanes 0-15, 1: lanes 16-31
  const int l    = lane & 15;
  const int rowBase = blockIdx.x * 64 + wid * 16;

  // A-row handled by this lane (ISA: 16x4 f32 A -> lanes carry M, VGPR pair + half carries K)
  int arow = rowBase + l;
  if (arow > M - 1) arow = M - 1;   // clamped load; store is guarded below

  float rs = 1.0f;
  if (RSMODE == 1) { rs = (deg[arow] > 0.0f) ? 1.0f : 0.0f; }
  if (RSMODE == 2) { float d = deg[arow]; rs = (d > 0.0f) ? (1.0f / d) : 0.0f; }

  v8f acc[NOUT / 16];
  #pragma unroll
  for (int nt = 0; nt < NOUT / 16; ++nt) acc[nt] = v8f{};

  #pragma unroll 4
  for (int kk = 0; kk < K; kk += 4) {
    const int kb = kk + half * 2;
    v2f a;
    a.x = A[arow * K + kb + 0] * rs;
    a.y = A[arow * K + kb + 1] * rs;
    #pragma unroll
    for (int nt = 0; nt < NOUT / 16; ++nt) {
      const int col = nt * 16 + l;   // B: lanes carry N, VGPR pair + half carries K
      v2f b;
      b.x = sW[(kb + 0) * NOUT + col];
      b.y = sW[(kb + 1) * NOUT + col];
      acc[nt] = __builtin_amdgcn_wmma_f32_16x16x4_f32(
          /*neg_a=*/false, a, /*neg_b=*/false, b,
          /*c_mod=*/(short)0, acc[nt], /*reuse_a=*/false, /*reuse_b=*/false);
    }
  }

  // C/D layout: VGPR r -> row (r + half*8), lane l -> column within tile.
  #pragma unroll
  for (int nt = 0; nt < NOUT / 16; ++nt) {
    const int col = nt * 16 + l;
    #pragma unroll
    for (int r = 0; r < 8; ++r) {
      const int row = rowBase + half * 8 + r;
      if (row < M) {
        float v = acc[nt][r];
        if (ACCUM) v += out[row * NOUT + col];
        if (BIAS)  v += bias[col];
        if (RELU)  v = v > 0.0f ? v : 0.0f;
        out[row * NOUT + col] = v;
      }
    }
  }
}

// ---------------------------------------------------------------------------
// Sparse / elementwise kernels
// ---------------------------------------------------------------------------
__global__ void count_kernel(const int* __restrict__ src, const int* __restrict__ dst,
                             float* __restrict__ degV, float* __restrict__ cntE, int nnz) {
  int i = blockIdx.x * blockDim.x + threadIdx.x;
  if (i < nnz) {
    atomicAdd(&degV[src[i]], 1.0f);
    atomicAdd(&cntE[dst[i]], 1.0f);
  }
}

// XeSum[dst] += h[src]   (64 feature channels)
__global__ void scatter_v2e_kernel(const float* __restrict__ h,
                                   const int* __restrict__ src, const int* __restrict__ dst,
                                   float* __restrict__ XeSum, int total) {
  int idx = blockIdx.x * blockDim.x + threadIdx.x;
  if (idx < total) {
    int n = idx >> 6, c = idx & 63;
    atomicAdd(&XeSum[dst[n] * 64 + c], h[src[n] * 64 + c]);
  }
}

// ZSum[src] += XeSum[dst] / max(cntE[dst],1)   (edge mean folded in)
__global__ void scatter_e2v_kernel(const float* __restrict__ XeSum, const float* __restrict__ cntE,
                                   const int* __restrict__ src, const int* __restrict__ dst,
                                   float* __restrict__ ZSum, int total) {
  int idx = blockIdx.x * blockDim.x + threadIdx.x;
  if (idx < total) {
    int n = idx >> 6, c = idx & 63;
    int e = dst[n];
    float cnt = cntE[e];
    float inv = cnt > 0.0f ? 1.0f / cnt : 0.0f;
    atomicAdd(&ZSum[src[n] * 64 + c], XeSum[e * 64 + c] * inv);
  }
}

// V = 0.5*(P + mask*b2) + 0.5*x0
__global__ void combine_kernel(const float* __restrict__ P, const float* __restrict__ b2,
                               const float* __restrict__ deg, const float* __restrict__ x0,
                               float* __restrict__ V, int total) {
  int idx = blockIdx.x * blockDim.x + threadIdx.x;
  if (idx < total) {
    int i = idx >> 6, c = idx & 63;
    float m = deg[i] > 0.0f ? 1.0f : 0.0f;
    V[idx] = 0.5f * (P[idx] + m * b2[c]) + 0.5f * x0[idx];
  }
}

__global__ void countB_kernel(const int* __restrict__ batch, float* __restrict__ cntB, int n) {
  int i = blockIdx.x * blockDim.x + threadIdx.x;
  if (i < n) atomicAdd(&cntB[batch[i]], 1.0f);
}

__global__ void scatter_out_kernel(const float* __restrict__ logits, const int* __restrict__ batch,
                                   float* __restrict__ outSum, int total) {
  int idx = blockIdx.x * blockDim.x + threadIdx.x;
  if (idx < total) {
    int i = idx >> 5, c = idx & 31;
    atomicAdd(&outSum[batch[i] * 32 + c], logits[idx]);
  }
}

__global__ void finalize_kernel(const float* __restrict__ outSum, const float* __restrict__ cntB,
                                float* __restrict__ out, int total) {
  int idx = blockIdx.x * blockDim.x + threadIdx.x;
  if (idx < total) {
    float c = cntB[idx >> 5];
    out[idx] = outSum[idx] / fmaxf(c, 1.0f);
  }
}

// ---------------------------------------------------------------------------
extern "C" void kernel_launch(void* const* d_in, const int* in_sizes, int n_in,
                              void* d_out, int out_size, void* d_ws, size_t ws_size,
                              hipStream_t stream) {
  const float* X        = (const float*)d_in[0];
  const int*   v2e_src  = (const int*)d_in[1];
  const int*   v2e_dst  = (const int*)d_in[2];
  const int*   allb     = (const int*)d_in[3];
  const float* W_in = (const float*)d_in[4];  const float* b_in = (const float*)d_in[5];
  const float* W1a  = (const float*)d_in[6];  const float* b1a  = (const float*)d_in[7];
  const float* W1b  = (const float*)d_in[8];  const float* b1b  = (const float*)d_in[9];
  const float* W2   = (const float*)d_in[10]; const float* b2   = (const float*)d_in[11];
  const float* W3   = (const float*)d_in[12]; const float* b3   = (const float*)d_in[13];
  const float* Wc1  = (const float*)d_in[14]; const float* bc1  = (const float*)d_in[15];
  const float* Wc2  = (const float*)d_in[16]; const float* bc2  = (const float*)d_in[17];

  const int Nn = NN_NODES, Ee = EE_EDGES, nnz = NNZ_INC;

  // Workspace carve-up
  char* p = (char*)d_ws;
  float* bufX  = (float*)p; p += (size_t)Nn * 64 * 4;   // current x
  float* bufX0 = (float*)p; p += (size_t)Nn * 64 * 4;   // x0 residual
  float* bufT  = (float*)p; p += (size_t)Nn * 64 * 4;   // t/h, ZSum, V
  float* bufH  = (float*)p; p += (size_t)Nn * 64 * 4;   // P, cls logits
  float* bufXe = (float*)p; p += (size_t)Ee * 64 * 4;   // XeSum
  float* degV  = (float*)p; p += (size_t)Nn * 4;
  float* cntE  = (float*)p; p += (size_t)Ee * 4;        // contiguous with degV
  float* outSum= (float*)p; p += (size_t)NGR_DIM * 32 * 4;
  float* cntB  = (float*)p; p += (size_t)NGR_DIM * 4;   // contiguous with outSum

  const dim3 blk(128);
  const int gM = (Nn + 63) / 64;                 // row tiles for GEMMs
  const int nnzT = nnz * 64;

  // Degrees / edge sizes (degV and cntE are adjacent)
  hipMemsetAsync(degV, 0, (size_t)(Nn + Ee) * 4, stream);
  count_kernel<<<(nnz + 255) / 256, 256, 0, stream>>>(v2e_src, v2e_dst, degV, cntE, nnz);

  // x = relu(X @ W_in + b_in)
  gemm_wmma_kernel<FT_DIM, 64, true, true, false, 0><<<gM, blk, 0, stream>>>(X, W_in, b_in, nullptr, bufX, Nn);
  hipMemcpyAsync(bufX0, bufX, (size_t)Nn * 64 * 4, hipMemcpyDeviceToDevice, stream);

  for (int layer = 0; layer < 2; ++layer) {
    // h = relu(x@W1a+b1a)@W1b+b1b     (second GEMM in-place: tile reads precede stores)
    gemm_wmma_kernel<64, 64, true,  true, false, 0><<<gM, blk, 0, stream>>>(bufX, W1a, b1a, nullptr, bufT, Nn);
    gemm_wmma_kernel<64, 64, false, true, false, 0><<<gM, blk, 0, stream>>>(bufT, W1b, b1b, nullptr, bufT, Nn);

    // XeSum[e] = sum_{src in e} h[src]
    hipMemsetAsync(bufXe, 0, (size_t)Ee * 64 * 4, stream);
    scatter_v2e_kernel<<<(nnzT + 255) / 256, 256, 0, stream>>>(bufT, v2e_src, v2e_dst, bufXe, nnzT);

    // ZSum[i] = sum_{e: i in e} XeSum[e]/|e|
    hipMemsetAsync(bufT, 0, (size_t)Nn * 64 * 4, stream);
    scatter_e2v_kernel<<<(nnzT + 255) / 256, 256, 0, stream>>>(bufXe, cntE, v2e_src, v2e_dst, bufT, nnzT);

    // P = mask(x) @ W2_top  +  (ZSum/deg) @ W2_bot      (row scales fused into GEMM)
    gemm_wmma_kernel<64, 64, false, false, false, 1><<<gM, blk, 0, stream>>>(bufX, W2,           nullptr, degV, bufH, Nn);
    gemm_wmma_kernel<64, 64, false, false, true,  2><<<gM, blk, 0, stream>>>(bufT, W2 + 64 * 64, nullptr, degV, bufH, Nn);

    // V = 0.5*(P + mask*b2) + 0.5*x0 ; x = relu(V@W3+b3)
    combine_kernel<<<(Nn * 64 + 255) / 256, 256, 0, stream>>>(bufH, b2, degV, bufX0, bufT, Nn * 64);
    gemm_wmma_kernel<64, 64, true, true, false, 0><<<gM, blk, 0, stream>>>(bufT, W3, b3, nullptr, bufX, Nn);
  }

  // Classifier: logits = relu(x@Wc1+bc1)@Wc2+bc2
  gemm_wmma_kernel<64, 64,       true,  true, false, 0><<<gM, blk, 0, stream>>>(bufX, Wc1, bc1, nullptr, bufT, Nn);
  gemm_wmma_kernel<64, NCLS_DIM, false, true, false, 0><<<gM, blk, 0, stream>>>(bufT, Wc2, bc2, nullptr, bufH, Nn);

  // Per-graph mean readout
  hipMemsetAsync(outSum, 0, (size_t)(NGR_DIM * 32 + NGR_DIM) * 4, stream);
  countB_kernel<<<(Nn + 255) / 256, 256, 0, stream>>>(allb, cntB, Nn);
  scatter_out_kernel<<<(Nn * 32 + 255) / 256, 256, 0, stream>>>(bufH, allb, outSum, Nn * 32);
  finalize_kernel<<<(NGR_DIM * 32 + 255) / 256, 256, 0, stream>>>(outSum, cntB, (float*)d_out, NGR_DIM * 32);
}